// PriorLayer_42623255446174
// MI455X (gfx1250) — compile-verified
//
#include <hip/hip_runtime.h>
#include <hip/hip_bf16.h>

// ---------------------------------------------------------------------------
// PriorLayer online sum-product message passing, restructured as a blocked
// associative scan over 64x64 transfer matrices so the MI455X's WMMA units
// (V_WMMA_F32_16X16X4_F32) do the heavy lifting instead of a 262144-long
// serial matvec chain.
// ---------------------------------------------------------------------------

#define DIM      64
#define L_CHUNK  1024
#define N_CHUNK  256
#define T_TOTAL  262144   // N_CHUNK * L_CHUNK

typedef __attribute__((ext_vector_type(2))) float v2f;
typedef __attribute__((ext_vector_type(8))) float v8f;

// ---------------------------------------------------------------------------
// Phase A: per-chunk transfer-matrix products  P_c = prod_{t in chunk} (D_t T)
// One block per chunk, 512 threads = 16 waves; wave w owns output tile
// (ti = w/4, tj = w%4). Constant T fragments live in registers; P double-
// buffers in LDS; per-step sum renormalization keeps fp32 in range (the scale
// cancels later).
// ---------------------------------------------------------------------------
__global__ __launch_bounds__(512)
void phaseA_chunk_products(const float* __restrict__ ps,
                           const float* __restrict__ Tm,
                           float* __restrict__ chunkP)
{
    __shared__ float buf[2][DIM * DIM];   // 32 KB double-buffered P
    __shared__ float ps_row[DIM];
    __shared__ float wsum[16];

    const int tid  = threadIdx.x;
    const int lane = tid & 31;
    const int wave = tid >> 5;
    const int ti   = wave >> 2;
    const int tj   = wave & 3;
    const int half = lane >> 4;             // 0: lanes 0-15, 1: lanes 16-31
    const int col  = 16 * tj + (lane & 15); // output column (B/C/D N index)
    const int chunk = blockIdx.x;
    const size_t base = (size_t)chunk * L_CHUNK;

    // Constant A fragments of T for this wave's tile row (16x64 strip).
    // A 16x4 f32 layout: lanes 0-15 M=0..15, VGPR0 holds K=(half*2),
    // VGPR1 holds K=(half*2+1), per 4-wide K-block kb.
    v2f afrag[16];
    {
        const int arow = 16 * ti + (lane & 15);
        #pragma unroll
        for (int kb = 0; kb < 16; ++kb) {
            const int k0 = 4 * kb + 2 * half;
            afrag[kb].x = Tm[arow * DIM + k0];
            afrag[kb].y = Tm[arow * DIM + k0 + 1];
        }
    }

    // P = Identity, written through the C/D ownership map (covers all 4096).
    #pragma unroll
    for (int v = 0; v < 8; ++v) {
        const int r = 16 * ti + v + 8 * half;
        buf[0][r * DIM + col] = (r == col) ? 1.0f : 0.0f;
    }
    if (tid < DIM) ps_row[tid] = ps[base * DIM + tid];
    __syncthreads();

    for (int t = 0; t < L_CHUNK; ++t) {
        const int curp = t & 1;

        // TP = T @ P  (this wave's 16x16 tile), K = 64 in 16 steps of 4.
        v8f acc = {};
        #pragma unroll
        for (int kb = 0; kb < 16; ++kb) {
            const int krow = 4 * kb + 2 * half;
            v2f b;
            b.x = buf[curp][(krow    ) * DIM + col];
            b.y = buf[curp][(krow + 1) * DIM + col];
            acc = __builtin_amdgcn_wmma_f32_16x16x4_f32(
                      /*neg_a=*/false, afrag[kb],
                      /*neg_b=*/false, b,
                      /*c_mod=*/(short)0, acc,
                      /*reuse_a=*/false, /*reuse_b=*/false);
        }

        // Row scaling by ps_t and partial sum for renormalization.
        float scaled[8];
        float part = 0.0f;
        #pragma unroll
        for (int v = 0; v < 8; ++v) {
            const int r = 16 * ti + v + 8 * half;
            const float x = acc[v] * ps_row[r];
            scaled[v] = x;
            part += x;
        }
        // Deterministic reduction: wave32 shuffle tree, then 16-wave LDS sum.
        #pragma unroll
        for (int off = 16; off > 0; off >>= 1)
            part += __shfl_xor(part, off, 32);
        if (lane == 0) wsum[wave] = part;
        __syncthreads();

        float total = 0.0f;
        #pragma unroll
        for (int w = 0; w < 16; ++w) total += wsum[w];
        const float inv = 1.0f / total;

        #pragma unroll
        for (int v = 0; v < 8; ++v) {
            const int r = 16 * ti + v + 8 * half;
            buf[curp ^ 1][r * DIM + col] = scaled[v] * inv;
        }
        if (tid < DIM && (t + 1) < L_CHUNK)
            ps_row[tid] = ps[(base + t + 1) * DIM + tid];
        __syncthreads();
    }

    // L_CHUNK is even -> final matrix sits in buf[0].
    float* out = chunkP + (size_t)chunk * DIM * DIM;
    for (int i = tid; i < DIM * DIM; i += 512) out[i] = buf[0][i];
}

// ---------------------------------------------------------------------------
// Phase B: tiny serial scan over the 256 chunk matrices; records the
// normalized state at each chunk entry.
// ---------------------------------------------------------------------------
__global__ __launch_bounds__(64)
void phaseB_scan(const float* __restrict__ chunkP,
                 const float* __restrict__ state0,
                 float* __restrict__ chunkStates)
{
    __shared__ float s[DIM];
    __shared__ float red[2];
    const int r = threadIdx.x;
    const int lane = r & 31;
    s[r] = state0[r];
    __syncthreads();

    for (int c = 0; c < N_CHUNK; ++c) {
        chunkStates[c * DIM + r] = s[r];
        const float* P = chunkP + (size_t)c * DIM * DIM;
        float acc = 0.0f;
        #pragma unroll 8
        for (int k = 0; k < DIM; ++k) acc += P[r * DIM + k] * s[k];

        float part = acc;
        #pragma unroll
        for (int off = 16; off > 0; off >>= 1)
            part += __shfl_xor(part, off, 32);
        if (lane == 0) red[r >> 5] = part;
        __syncthreads();
        const float total = red[0] + red[1];
        __syncthreads();
        s[r] = acc / total;
        __syncthreads();
    }
}

// ---------------------------------------------------------------------------
// Phase C: parallel per-chunk replay exactly matching the reference step,
// emitting E[HR] and entropy per timestep.
// ---------------------------------------------------------------------------
__global__ __launch_bounds__(64)
void phaseC_emit(const float* __restrict__ ps,
                 const float* __restrict__ Tm,
                 const float* __restrict__ bins,
                 const float* __restrict__ chunkStates,
                 float* __restrict__ outE,
                 float* __restrict__ outU)
{
    __shared__ float Tl[DIM * DIM];
    __shared__ float sbuf[2][DIM];
    __shared__ float redN[2];
    __shared__ float redE[2];
    __shared__ float redU[2];

    const int r = threadIdx.x;        // 0..63, one matrix row per thread
    const int lane = r & 31;
    const int wave = r >> 5;
    const int chunk = blockIdx.x;
    const size_t base = (size_t)chunk * L_CHUNK;

    for (int i = r; i < DIM * DIM; i += 64) Tl[i] = Tm[i];
    sbuf[0][r] = chunkStates[chunk * DIM + r];
    const float binr = bins[r];
    __syncthreads();

    for (int t = 0; t < L_CHUNK; ++t) {
        const int curp = t & 1;
        const size_t g = base + t;

        float pr = 0.0f;
        #pragma unroll 8
        for (int k = 0; k < DIM; ++k) pr += Tl[r * DIM + k] * sbuf[curp][k];
        const float p = pr * ps[g * DIM + r];

        float part = p;
        #pragma unroll
        for (int off = 16; off > 0; off >>= 1)
            part += __shfl_xor(part, off, 32);
        if (lane == 0) redN[wave] = part;
        __syncthreads();

        const float total = redN[0] + redN[1];
        const float prob = p / total;
        sbuf[curp ^ 1][r] = prob;

        float e = prob * binr;
        float u = -prob * __logf(prob + 1e-10f);
        #pragma unroll
        for (int off = 16; off > 0; off >>= 1) {
            e += __shfl_xor(e, off, 32);
            u += __shfl_xor(u, off, 32);
        }
        if (lane == 0) { redE[wave] = e; redU[wave] = u; }
        __syncthreads();

        if (r == 0) {
            outE[g] = redE[0] + redE[1];
            outU[g] = redU[0] + redU[1];
        }
    }
}

// ---------------------------------------------------------------------------
extern "C" void kernel_launch(void* const* d_in, const int* in_sizes, int n_in,
                              void* d_out, int out_size, void* d_ws, size_t ws_size,
                              hipStream_t stream) {
    const float* ps    = (const float*)d_in[0];   // (T, 64)
    const float* Tm    = (const float*)d_in[1];   // (64, 64)
    const float* bins  = (const float*)d_in[2];   // (64,)
    const float* state = (const float*)d_in[3];   // (64,)

    float* outE = (float*)d_out;                  // E_x  : T floats
    float* outU = (float*)d_out + T_TOTAL;        // uncert: T floats

    float* chunkP      = (float*)d_ws;                              // 256*4096 f32
    float* chunkStates = chunkP + (size_t)N_CHUNK * DIM * DIM;      // 256*64  f32

    phaseA_chunk_products<<<N_CHUNK, 512, 0, stream>>>(ps, Tm, chunkP);
    phaseB_scan<<<1, 64, 0, stream>>>(chunkP, state, chunkStates);
    phaseC_emit<<<N_CHUNK, 64, 0, stream>>>(ps, Tm, bins, chunkStates, outE, outU);
}